// analyse_33638183862871
// MI455X (gfx1250) — compile-verified
//
#include <hip/hip_runtime.h>
#include <hip/hip_bf16.h>

typedef __attribute__((ext_vector_type(2))) float v2f;
typedef __attribute__((ext_vector_type(8))) float v8f;
typedef __attribute__((ext_vector_type(4))) unsigned int u32x4;
typedef __attribute__((ext_vector_type(4))) int i32x4;
typedef __attribute__((ext_vector_type(8))) int i32x8;

#define N_PTS 4096
#define NTHREADS 256
#define NWAVES (NTHREADS / 32)
#define CH 16

#if defined(__has_builtin)
#if __has_builtin(__builtin_amdgcn_tensor_load_to_lds) && \
    __has_builtin(__builtin_amdgcn_s_wait_tensorcnt)
#define HAVE_TDM 1
#endif
#endif

#ifdef HAVE_TDM
// ---------------------------------------------------------------------------
// Issue one TDM 2D tile load: copy tile (t0 x t1 elements, 4B each) from a
// row-major global tensor (row length td0, row stride s0, td1 rows) into LDS
// at lds_addr (rows packed back-to-back, t0*4 bytes apart).
// D# layout per CDNA5 ISA 8.3/8.4 (group0: flags/lds/global/type,
// group1: data_size + dims + dim0 stride). Groups 2/3 zero (<=2D tensor).
// ---------------------------------------------------------------------------
__device__ __forceinline__ void tdm_load_2d(unsigned lds_addr, const void* gptr,
                                            unsigned td0, unsigned td1,
                                            unsigned t0, unsigned t1,
                                            unsigned long long s0) {
    unsigned long long ga = (unsigned long long)gptr;
    u32x4 g0;
    g0[0] = 1u;                                        // count=1, no gather
    g0[1] = lds_addr;                                  // LDS byte address
    g0[2] = (unsigned)(ga & 0xFFFFFFFFu);              // global_addr[31:0]
    g0[3] = (unsigned)((ga >> 32) & 0x01FFFFFFu)       // global_addr[56:32]
            | (2u << 30);                              // type=2 ("image")
    i32x8 g1;
    g1[0] = (int)(2u << 16);                           // data_size=2 -> 4B
    g1[1] = (int)((td0 & 0xFFFFu) << 16);              // tensor_dim0[15:0]
    g1[2] = (int)(((td0 >> 16) & 0xFFFFu) | ((td1 & 0xFFFFu) << 16));
    g1[3] = (int)(((td1 >> 16) & 0xFFFFu) | ((t0 & 0xFFFFu) << 16));
    g1[4] = (int)(t1 & 0xFFFFu);                       // tile_dim1 (tile_dim2=0)
    g1[5] = (int)(s0 & 0xFFFFFFFFull);                 // dim0_stride[31:0]
    g1[6] = (int)((s0 >> 32) & 0xFFFFull);             // dim0_stride[47:32]
    g1[7] = 0;
    i32x4 z4 = {0, 0, 0, 0};
#if __has_include(<hip/amd_detail/amd_gfx1250_TDM.h>)
    i32x8 z8 = {0, 0, 0, 0, 0, 0, 0, 0};
    __builtin_amdgcn_tensor_load_to_lds(g0, g1, z4, z4, z8, 0);   // clang-23 form
#else
    __builtin_amdgcn_tensor_load_to_lds(g0, g1, z4, z4, 0);       // ROCm 7.2 form
#endif
}
#endif // HAVE_TDM

// lexicographic (key asc, idx asc) "greater-than"
__device__ __forceinline__ bool kv_greater(float ka, float ia, float kb, float ib) {
    return (ka > kb) || ((ka == kb) && (ia > ib));
}

// ---------------------------------------------------------------------------
// Kernel A: organize + stable sort by conf + greedy NMS (WMMA f32 16x16x4
// distance tiles). One block per sequence: bid = bb*4 + e*2 + w.
// Writes [z,x,y,keep,0] per point into d_out.
// ---------------------------------------------------------------------------
__global__ __launch_bounds__(NTHREADS) void nms_sort_kernel(
    const float* __restrict__ pred, const float* __restrict__ targ,
    float* __restrict__ out) {
    __shared__ float sC1[N_PTS];          // key during sort, then z coord
    __shared__ float sC2[N_PTS];          // idx (as float) during sort, then x
    __shared__ float sC3[N_PTS];          // conf staging, then y coord
    __shared__ unsigned char sKeep[N_PTS];
    __shared__ float sD2[CH * CH];        // intra-chunk d2 tile
    __shared__ unsigned int sSup;         // per-chunk suppression bits

    const int tid = threadIdx.x;
    const int bid = blockIdx.x;           // 0..15
    const int w   = bid & 1;              // 0=pred, 1=targ
    const int e   = (bid >> 1) & 1;       // element 0/1
    const int bb  = bid >> 2;             // batch
    const float* src = (w == 0) ? pred : targ;

    const float rr = (e == 0) ? (float)(0.74 * 1.4) : (float)(0.528 * 1.4);
    const float r2 = rr * rr;
    const float INF = __builtin_inff();

    const float* confp = src + ((size_t)bb * 8 + e) * N_PTS;

#ifdef HAVE_TDM
    // ---- phase 0: TDM: DMA the contiguous conf row into LDS (sC3 is free
    // until phase 3). Tensor ops ignore EXEC; only wave 0 reaches this branch.
    if (tid == 0) {
        tdm_load_2d((unsigned)(unsigned long long)(void*)&sC3[0], confp,
                    /*td0=*/N_PTS, /*td1=*/1, /*t0=*/N_PTS, /*t1=*/1,
                    /*s0=*/N_PTS);
        __builtin_amdgcn_s_wait_tensorcnt(0);
    }
    __syncthreads();
#endif

    // ---- phase 1: sort keys: -conf for valid, +inf for invalid; idx payload
    for (int n = tid; n < N_PTS; n += NTHREADS) {
#ifdef HAVE_TDM
        float conf = sC3[n];
#else
        float conf = confp[n];
#endif
        sC1[n] = (conf > 0.5f) ? -conf : INF;
        sC2[n] = (float)n;                 // exact for n < 2^24
    }
    __syncthreads();

    // ---- phase 2: bitonic sort ascending by (key, idx) -> stable argsort
    for (int k = 2; k <= N_PTS; k <<= 1) {
        for (int j = k >> 1; j > 0; j >>= 1) {
            for (int i = tid; i < N_PTS; i += NTHREADS) {
                int ixj = i ^ j;
                if (ixj > i) {
                    float ka = sC1[i],   ia = sC2[i];
                    float kb = sC1[ixj], ib = sC2[ixj];
                    bool up = ((i & k) == 0);
                    bool g  = kv_greater(ka, ia, kb, ib);
                    if (g == up) {        // swap
                        sC1[i] = kb;   sC2[i] = ib;
                        sC1[ixj] = ka; sC2[ixj] = ia;
                    }
                }
            }
            __syncthreads();
        }
    }

    // ---- phase 3: gather organized coords into sorted order (in place)
    for (int n = tid; n < N_PTS; n += NTHREADS) {
        float key = sC1[n];
        int i = (int)sC2[n];
        int zi = i >> 10, xi = (i >> 5) & 31, yi = i & 31;
        size_t base = (size_t)bb * 8 * N_PTS;
        float zc = (src[base + (size_t)(2 + e) * N_PTS + i] + (float)zi) * 0.75f;
        float xc = (src[base + (size_t)(4 + e) * N_PTS + i] + (float)xi) * 0.78125f;
        float yc = (src[base + (size_t)(6 + e) * N_PTS + i] + (float)yi) * 0.78125f;
        sC1[n] = zc; sC2[n] = xc; sC3[n] = yc;     // same-thread RAW on n only
        sKeep[n] = (key < INF) ? 1 : 0;            // valid flag
    }
    __syncthreads();

    const int lane   = tid & 31;
    const int waveId = tid >> 5;
    const int half   = lane >> 4;      // 0: K0/K1 rows, 1: K2/K3 rows
    const int colM   = lane & 15;

    // ---- phase 4: greedy NMS in 16-row chunks.
    // d2 = |p|^2 + |q|^2 - 2 p.q computed as one WMMA:
    //   A row M = (-2z, -2x, -2y, |p|^2), B col N = (z, x, y, 1), C = |q|^2.
    for (int c = 0; c < N_PTS / CH; ++c) {
        const int c0 = c * CH;
        if (tid == 0) sSup = 0u;
        __syncthreads();

        // A fragment for this chunk (shared by all tiles this wave touches)
        int rI = c0 + colM;
        float p1 = sC1[rI], p2 = sC2[rI], p3 = sC3[rI];
        float pn2 = p1 * p1 + p2 * p2 + p3 * p3;
        v2f a;
        a.x = half ? (-2.0f * p3) : (-2.0f * p1);
        a.y = half ? pn2          : (-2.0f * p2);

        // cross-tile suppression vs all finalized earlier chunks
        unsigned int mask = 0u;
        for (int t = waveId; t < c; t += NWAVES) {   // wave-uniform trip count
            int q = t * CH + colM;
            float q1 = sC1[q], q2 = sC2[q], q3 = sC3[q];
            float qn2 = q1 * q1 + q2 * q2 + q3 * q3;
            v2f b;
            b.x = half ? q3   : q1;
            b.y = half ? 1.0f : q2;
            v8f cc;
#pragma unroll
            for (int m = 0; m < 8; ++m) cc[m] = qn2;
            v8f d = __builtin_amdgcn_wmma_f32_16x16x4_f32(
                false, a, false, b, (short)0, cc, false, false);
            bool kq = (sKeep[q] != 0);
#pragma unroll
            for (int m = 0; m < 8; ++m) {
                if (kq && d[m] < r2) mask |= (1u << (m + half * 8));
            }
        }
        if (mask) atomicOr(&sSup, mask);

        // wave 0: intra-chunk self d2 tile
        if (waveId == 0) {
            int q = c0 + colM;
            float q1 = sC1[q], q2 = sC2[q], q3 = sC3[q];
            float qn2 = q1 * q1 + q2 * q2 + q3 * q3;
            v2f b;
            b.x = half ? q3   : q1;
            b.y = half ? 1.0f : q2;
            v8f cc;
#pragma unroll
            for (int m = 0; m < 8; ++m) cc[m] = qn2;
            v8f d = __builtin_amdgcn_wmma_f32_16x16x4_f32(
                false, a, false, b, (short)0, cc, false, false);
#pragma unroll
            for (int m = 0; m < 8; ++m)
                sD2[(m + half * 8) * CH + colM] = d[m];
        }
        __syncthreads();

        // serial resolve of the 16-row recurrence (exactly the reference loop)
        if (tid == 0) {
            unsigned int sup0 = sSup;
            for (int r = 0; r < CH; ++r) {
                int n = c0 + r;
                if (sKeep[n]) {
                    bool sup = ((sup0 >> r) & 1u) != 0u;
                    for (int jj = 0; jj < r; ++jj)
                        if (sKeep[c0 + jj] && sD2[jj * CH + r] < r2) sup = true;
                    if (sup) sKeep[n] = 0;
                }
            }
        }
        __syncthreads();
    }

    // ---- phase 5: emit [z, x, y, keep, matched=0]; seq index == bid
    float* obase = out + (size_t)bid * N_PTS * 5;
    for (int n = tid; n < N_PTS; n += NTHREADS) {
        size_t o = (size_t)n * 5;
        obase[o + 0] = sC1[n];
        obase[o + 1] = sC2[n];
        obase[o + 2] = sC3[n];
        obase[o + 3] = sKeep[n] ? 1.0f : 0.0f;
        obase[o + 4] = 0.0f;
    }
}

// ---------------------------------------------------------------------------
// Kernel B: greedy first-fit matching per (b, e). One block per pair; wave 0
// runs the serial-by-construction loop, lanes scan 32 targets at a time and
// ballot/ffs picks the FIRST (lowest sorted index) available target.
// Target coords staged in LDS by a TDM 2D tile load (3-of-5 floats per row).
// ---------------------------------------------------------------------------
__global__ __launch_bounds__(NTHREADS) void match_kernel(float* __restrict__ out) {
    __shared__ float tS[N_PTS * 3];       // packed 12B rows: z,x,y per target
    __shared__ unsigned char kt[N_PTS];
    __shared__ unsigned char mt[N_PTS];

    const int tid = threadIdx.x;
    const int bid = blockIdx.x;           // 0..7
    const int e  = bid & 1;
    const int bb = bid >> 1;
    const float rr = (e == 0) ? (float)(0.74 * 1.4) : (float)(0.528 * 1.4);
    const float r2 = rr * rr;

    float* pbase = out + (size_t)((bb * 2 + e) * 2 + 0) * N_PTS * 5;
    float* tbase = pbase + (size_t)N_PTS * 5;

#ifdef HAVE_TDM
    // TDM: tile_dim0=3 of a row of td0=5 floats, stride 5, 4096 rows.
    if (tid == 0) {
        tdm_load_2d((unsigned)(unsigned long long)(void*)&tS[0], tbase,
                    /*td0=*/5, /*td1=*/N_PTS, /*t0=*/3, /*t1=*/N_PTS, /*s0=*/5);
        __builtin_amdgcn_s_wait_tensorcnt(0);
    }
#endif
    for (int n = tid; n < N_PTS; n += NTHREADS) {
        size_t o = (size_t)n * 5;
#ifndef HAVE_TDM
        tS[n * 3 + 0] = tbase[o + 0];
        tS[n * 3 + 1] = tbase[o + 1];
        tS[n * 3 + 2] = tbase[o + 2];
#endif
        kt[n] = (tbase[o + 3] > 0.5f) ? 1 : 0;
        mt[n] = 0;
    }
    __syncthreads();

    if (tid < 32) {
        const int lane = tid;
        volatile unsigned char* mtv = mt;
        for (int a = 0; a < N_PTS; ++a) {
            size_t po = (size_t)a * 5;
            __builtin_prefetch(pbase + po + 5, 0, 1);   // next pred row
            bool kp = pbase[po + 3] > 0.5f;   // uniform across the wave
            float mp = 0.0f;
            if (kp) {
                float p1 = pbase[po + 0], p2 = pbase[po + 1], p3 = pbase[po + 2];
                for (int j0 = 0; j0 < N_PTS; j0 += 32) {
                    int j = j0 + lane;
                    bool cand = (kt[j] != 0) && (mtv[j] == 0);
                    if (cand) {
                        float d1 = p1 - tS[j * 3 + 0];
                        float d2v = p2 - tS[j * 3 + 1];
                        float d3 = p3 - tS[j * 3 + 2];
                        cand = (d1 * d1 + d2v * d2v + d3 * d3) < r2;
                    }
                    unsigned int bal = __builtin_amdgcn_ballot_w32(cand);
                    if (bal) {                 // uniform: break together
                        int fb = __ffs(bal) - 1;
                        if (lane == fb) mtv[j] = 1;
                        mp = 1.0f;
                        break;
                    }
                }
            }
            if (lane == 0) pbase[po + 4] = mp;
        }
    }
    __syncthreads();

    for (int n = tid; n < N_PTS; n += NTHREADS)
        tbase[(size_t)n * 5 + 4] = mt[n] ? 1.0f : 0.0f;
}

// ---------------------------------------------------------------------------
extern "C" void kernel_launch(void* const* d_in, const int* in_sizes, int n_in,
                              void* d_out, int out_size, void* d_ws, size_t ws_size,
                              hipStream_t stream) {
    const float* pred = (const float*)d_in[0];
    const float* targ = (const float*)d_in[1];
    float* out = (float*)d_out;
    (void)in_sizes; (void)n_in; (void)out_size; (void)d_ws; (void)ws_size;

    nms_sort_kernel<<<dim3(16), dim3(NTHREADS), 0, stream>>>(pred, targ, out);
    match_kernel<<<dim3(8), dim3(NTHREADS), 0, stream>>>(out);
}